// QueryEncDec_81535659147664
// MI455X (gfx1250) — compile-verified
//
#include <hip/hip_runtime.h>
#include <hip/hip_bf16.h>

#define T_LEN   256   // timesteps
#define L_HALF  128   // layers per stack (encoder / decoder)
#define NLAYERS 256   // encoder + decoder chained

// ---- CDNA5 transcendental: v_tanh_f32 if available, exp-based fallback ----
__device__ __forceinline__ float fast_tanh(float x) {
#if __has_builtin(__builtin_amdgcn_tanhf)
  return __builtin_amdgcn_tanhf(x);
#else
  float e = __expf(2.0f * x);                       // v_exp_f32
  return (e - 1.0f) * __builtin_amdgcn_rcpf(e + 1.0f);
#endif
}

__device__ __forceinline__ float fast_sigmoid(float x) {
  // sigmoid(x) = 0.5*tanh(x/2) + 0.5  -> single transcendental
  return __builtin_fmaf(0.5f, fast_tanh(0.5f * x), 0.5f);
}

// One workgroup = one WGP. Thread l owns global layer l (0..127 encoder,
// 128..255 decoder). Systolic anti-diagonal sweep: at step s, thread l
// processes t = s - l, consuming thread (l-1)'s output from step s-1 via
// double-buffered LDS. Critical path = T + NLAYERS - 1 = 511 steps.
__global__ __launch_bounds__(NLAYERS) void gru_systolic_kernel(
    const float* __restrict__ X,
    const float* __restrict__ enc_w_ih, const float* __restrict__ enc_w_hh,
    const float* __restrict__ enc_b_ih, const float* __restrict__ enc_b_hh,
    const float* __restrict__ dec_w_ih, const float* __restrict__ dec_w_hh,
    const float* __restrict__ dec_b_ih, const float* __restrict__ dec_b_hh,
    float* __restrict__ out)   // [0..255]=dec_out, [256..383]=dec_h
{
  __shared__ float xlds[T_LEN];
  __shared__ float pipe[2][NLAYERS];

  const int l = threadIdx.x;

  // ---- Stage X[T] into LDS; prefer gfx1250 async global->LDS path --------
#if __has_builtin(__builtin_amdgcn_global_load_async_to_lds_b32)
  typedef __attribute__((address_space(1))) int gas_int;   // global
  typedef __attribute__((address_space(3))) int lds_int;   // LDS
  __builtin_amdgcn_global_load_async_to_lds_b32(
      (gas_int*)(X + l),            // per-lane global source
      (lds_int*)(&xlds[l]),         // per-lane LDS destination
      /*offset=*/0, /*cpol=*/0);
  #if __has_builtin(__builtin_amdgcn_s_wait_asynccnt)
  __builtin_amdgcn_s_wait_asynccnt(0);
  #else
  asm volatile("s_wait_asynccnt 0" ::: "memory");
  #endif
#else
  xlds[l] = X[l];
#endif

  // ---- Per-layer scalar parameters into registers (12 floats) ------------
  const bool  dec = (l >= L_HALF);
  const int   ll  = dec ? (l - L_HALF) : l;
  const float* wi = (dec ? dec_w_ih : enc_w_ih) + 3 * ll;
  const float* wh = (dec ? dec_w_hh : enc_w_hh) + 3 * ll;
  const float* bi = (dec ? dec_b_ih : enc_b_ih) + 3 * ll;
  const float* bh = (dec ? dec_b_hh : enc_b_hh) + 3 * ll;
  const float wi0 = wi[0], wi1 = wi[1], wi2 = wi[2];
  const float wh0 = wh[0], wh1 = wh[1], wh2 = wh[2];
  const float bi0 = bi[0], bi1 = bi[1], bi2 = bi[2];
  const float bh0 = bh[0], bh1 = bh[1], bh2 = bh[2];

  float h = 0.0f;
  __syncthreads();   // xlds visible to all waves

  // ---- Systolic sweep ----------------------------------------------------
  #pragma unroll 1
  for (int s = 0; s < T_LEN + NLAYERS - 1; ++s) {
    const int t = s - l;
    if (t >= 0 && t < T_LEN) {
      const float x = (l == 0) ? xlds[t] : pipe[(s - 1) & 1][l - 1];

      // torch gate order r, z, n
      const float r = fast_sigmoid(__builtin_fmaf(wi0, x, bi0) +
                                   __builtin_fmaf(wh0, h, bh0));
      const float z = fast_sigmoid(__builtin_fmaf(wi1, x, bi1) +
                                   __builtin_fmaf(wh1, h, bh1));
      const float n = fast_tanh(__builtin_fmaf(wi2, x, bi2) +
                                r * __builtin_fmaf(wh2, h, bh2));
      h = __builtin_fmaf(1.0f - z, n, z * h);

      pipe[s & 1][l] = h;                          // feed layer l+1 next step

      if (l == NLAYERS - 1) out[t] = h;            // dec_out[t]
      if (t == T_LEN - 1 && dec) out[T_LEN + ll] = h;  // dec_h[layer]
    }
    __syncthreads();   // step s writes visible before step s+1 reads
  }
}

extern "C" void kernel_launch(void* const* d_in, const int* in_sizes, int n_in,
                              void* d_out, int out_size, void* d_ws, size_t ws_size,
                              hipStream_t stream) {
  (void)in_sizes; (void)n_in; (void)d_ws; (void)ws_size; (void)out_size;
  const float* X        = (const float*)d_in[0];
  const float* enc_w_ih = (const float*)d_in[1];
  const float* enc_w_hh = (const float*)d_in[2];
  const float* enc_b_ih = (const float*)d_in[3];
  const float* enc_b_hh = (const float*)d_in[4];
  const float* dec_w_ih = (const float*)d_in[5];
  const float* dec_w_hh = (const float*)d_in[6];
  const float* dec_b_ih = (const float*)d_in[7];
  const float* dec_b_hh = (const float*)d_in[8];
  float* out = (float*)d_out;

  gru_systolic_kernel<<<1, NLAYERS, 0, stream>>>(
      X, enc_w_ih, enc_w_hh, enc_b_ih, enc_b_hh,
      dec_w_ih, dec_w_hh, dec_b_ih, dec_b_hh, out);
}